// Ndpdti_60292750901748
// MI455X (gfx1250) — compile-verified
//
#include <hip/hip_runtime.h>
#include <cstdint>

// ODE integration of 5 tiny MLPs (1x3 -> 3x3 -> 3x1, tanh) per sample, 50 Euler steps.
// Compute-bound (~260 flop/byte, tanh-dominated): scalar wave32 VALU + gfx1250
// v_tanh_f32 is optimal. WMMA is deliberately not used: the only GEMM is 3x3 with a
// tanh between layers -> 14% WMMA utilization + per-iteration cross-lane relayout,
// strictly worse than VALU. CDNA5-specific paths used: tensor_load_to_lds (TDM)
// weight staging + s_wait_tensorcnt + LDS broadcast reads + v_tanh_f32.

#define ODE_STEPS 50
#define ODE_DT 0.02f

typedef uint32_t v4u __attribute__((ext_vector_type(4)));
typedef uint32_t v8u __attribute__((ext_vector_type(8)));

__device__ __forceinline__ float tanh_hw(float x) {
#if __has_builtin(__builtin_amdgcn_tanhf)
    return __builtin_amdgcn_tanhf(x);
#else
    float r;
    asm("v_tanh_f32 %0, %1" : "=v"(r) : "v"(x));
    return r;
#endif
}

// 1-D TDM load: nelem f32 from global -> LDS offset lds_off.
// D# layout per CDNA5 ISA ch.10.7 / CDNA5 ISA: group0 = {count, lds_addr,
// global_addr, type=2}, group1 = {data_size=4B, tensor_dim0, tile_dim0, stride}.
// VADDR2/3 omitted (<=2D form).
__device__ __forceinline__ void tdm_load_1d_f32(uint32_t lds_off, const void* gsrc,
                                                uint32_t nelem) {
    uint64_t ga = (uint64_t)(uintptr_t)gsrc;
    v4u g0;
    g0[0] = 1u;                                   // count=1, is_restore=0, gather off
    g0[1] = lds_off;                              // lds_addr (bytes)
    g0[2] = (uint32_t)ga;                         // global_addr[31:0]
    g0[3] = ((uint32_t)(ga >> 32) & 0x01FFFFFFu)  // global_addr[56:32]
            | (2u << 30);                         // type = 2 ("image")
    v8u g1;
    g1[0] = (2u << 16);                           // workgroup_mask=0, data_size=4B
    g1[1] = (nelem & 0xFFFFu) << 16;              // tensor_dim0[15:0]
    g1[2] = (nelem >> 16) | (1u << 16);           // tensor_dim0[31:16], tensor_dim1=1
    g1[3] = (nelem & 0xFFFFu) << 16;              // tile_dim0
    g1[4] = 0u;                                   // tile_dim1=0, tile_dim2=0
    g1[5] = nelem;                                // tensor_dim0_stride[31:0]
    g1[6] = 0u;
    g1[7] = 0u;
    asm volatile("tensor_load_to_lds %0, %1" :: "s"(g0), "s"(g1) : "memory");
}

__device__ __forceinline__ void wait_tensorcnt0() {
#if __has_builtin(__builtin_amdgcn_s_wait_tensorcnt)
    __builtin_amdgcn_s_wait_tensorcnt(0);
#else
    asm volatile("s_wait_tensorcnt 0x0" ::: "memory");
#endif
}

struct F3 { float a, b, c; };  // 4-byte aligned 12B packet -> b64+b32 loads

__global__ __launch_bounds__(256) void ode5_mlp_kernel(
    const float* __restrict__ x,
    const float* __restrict__ W1,     // (5,1,3)
    const float* __restrict__ W2,     // (5,3,3)
    const float* __restrict__ W3,     // (5,3,1)
    const float* __restrict__ bvec,   // (5,1,1)
    const float* __restrict__ inp_mean,
    const float* __restrict__ inp_sd,
    const float* __restrict__ out_mean,
    const float* __restrict__ out_sd,
    float* __restrict__ out,          // d1[0:B], d2[B:2B], d3[2B:3B]
    int n)
{
    // Weight staging: wsh[0..14]=W1, [15..59]=W2, [60..74]=W3, [75..79]=b
    __shared__ float wsh[80];
    if (threadIdx.x < 32) {  // one wave issues the DMAs (TDM is wave-level, EXEC-ignored)
        uint32_t base = (uint32_t)(uintptr_t)(&wsh[0]);
        tdm_load_1d_f32(base + 0u,   W1,   15u);
        tdm_load_1d_f32(base + 60u,  W2,   45u);
        tdm_load_1d_f32(base + 240u, W3,   15u);
        tdm_load_1d_f32(base + 300u, bvec,  5u);
        wait_tensorcnt0();
    }
    __syncthreads();

    int i = blockIdx.x * blockDim.x + threadIdx.x;
    if (i >= n) return;

    const float im  = inp_mean[0];
    const float isd = inp_sd[0];
    const float om  = out_mean[0];
    const float osd = out_sd[0];
    const float rsd = 1.0f / isd;

    const F3 xi = *(const F3*)(x + 3 * (size_t)i);
    const float t1 = (xi.a - im) * rsd;
    const float t2 = (xi.b - im) * rsd;
    const float t3 = (xi.c - im) * rsd;
    const float tsum = t1 + t2 + t3;

    float s0v[5];
    s0v[0] = t1;
    s0v[1] = t1 + t2;
    s0v[2] = tsum;
    s0v[3] = tsum * tsum;  // (t1+t2+t3)^2 == n4
    s0v[4] = t1 * t1 + t2 * t2 + t3 * t3 - t1 * t2 - t1 * t3 - t2 * t3;

    float N[5];
#pragma unroll
    for (int nn = 0; nn < 5; ++nn) {
        const float a10 = wsh[3 * nn + 0];
        const float a11 = wsh[3 * nn + 1];
        const float a12 = wsh[3 * nn + 2];
        float w[9];
#pragma unroll
        for (int k = 0; k < 9; ++k) w[k] = wsh[15 + 9 * nn + k];  // W2[nn][i][j] = w[3i+j]
        const float a30 = wsh[60 + 3 * nn + 0];
        const float a31 = wsh[60 + 3 * nn + 1];
        const float a32 = wsh[60 + 3 * nn + 2];
        const float ebn = expf(wsh[75 + nn]);

        float s = s0v[nn];
#pragma unroll 2
        for (int step = 0; step < ODE_STEPS; ++step) {
            const float h0 = tanh_hw(s * a10);
            const float h1 = tanh_hw(s * a11);
            const float h2 = tanh_hw(s * a12);
            const float g0 = tanh_hw(fmaf(h2, w[6], fmaf(h1, w[3], h0 * w[0])));
            const float g1 = tanh_hw(fmaf(h2, w[7], fmaf(h1, w[4], h0 * w[1])));
            const float g2 = tanh_hw(fmaf(h2, w[8], fmaf(h1, w[5], h0 * w[2])));
            const float f  = fmaf(g2, a32, fmaf(g1, a31, fmaf(g0, a30, ebn)));
            s = fmaf(ODE_DT, f, s);
        }
        N[nn] = s;
    }

    const float c = 2.0f * N[3] * tsum;
    const float d1 = N[0] + N[1] + N[2] + c + N[4] * (2.0f * t1 - t2 - t3);
    const float d2 = N[1] + N[2] + c + N[4] * (2.0f * t2 - t1 - t3);
    const float d3 = N[2] + c + N[4] * (2.0f * t3 - t1 - t2);

    out[i]         = fmaf(d1, osd, om);
    out[n + i]     = fmaf(d2, osd, om);
    out[2 * n + i] = fmaf(d3, osd, om);
}

extern "C" void kernel_launch(void* const* d_in, const int* in_sizes, int n_in,
                              void* d_out, int out_size, void* d_ws, size_t ws_size,
                              hipStream_t stream) {
    (void)n_in; (void)d_ws; (void)ws_size; (void)out_size;
    const float* x        = (const float*)d_in[0];
    const float* W1       = (const float*)d_in[1];
    const float* W2       = (const float*)d_in[2];
    const float* W3       = (const float*)d_in[3];
    const float* bvec     = (const float*)d_in[4];
    const float* inp_mean = (const float*)d_in[5];
    const float* inp_sd   = (const float*)d_in[6];
    const float* out_mean = (const float*)d_in[7];
    const float* out_sd   = (const float*)d_in[8];
    float* out = (float*)d_out;

    const int n = in_sizes[0] / 3;  // B = 262144
    const int threads = 256;
    const int blocks = (n + threads - 1) / threads;
    ode5_mlp_kernel<<<blocks, threads, 0, stream>>>(
        x, W1, W2, W3, bvec, inp_mean, inp_sd, out_mean, out_sd, out, n);
}